// FQEMAVectorQuantizer_3624952398757
// MI455X (gfx1250) — compile-verified
//
#include <hip/hip_runtime.h>
#include <hip/hip_bf16.h>

typedef __attribute__((ext_vector_type(2))) float v2f;
typedef __attribute__((ext_vector_type(8))) float v8f;

#define NE        4096
#define EDIM      256
#define NTOK      32768
#define CHUNK     128          // codes staged in LDS per iteration
#define LDS_PITCH 260          // 256 + 4 pad floats -> conflict-free b64 reads
#define DECAY     0.99f
#define EPSV      1e-5f

// ---------------------------------------------------------------------------
// Kernel 0: per-code squared norms  e2[n] = sum_k e[n][k]^2   (one wave / row)
// ---------------------------------------------------------------------------
__global__ __launch_bounds__(256) void vq_e2_kernel(const float* __restrict__ emb,
                                                    float* __restrict__ e2) {
  int row = blockIdx.x * 8 + (threadIdx.x >> 5);
  int ln  = threadIdx.x & 31;
  const float* p = emb + (size_t)row * EDIM;
  float s = 0.f;
  #pragma unroll
  for (int k = ln; k < EDIM; k += 32) { float v = p[k]; s = fmaf(v, v, s); }
  #pragma unroll
  for (int m = 16; m >= 1; m >>= 1) s += __shfl_xor(s, m, 32);
  if (ln == 0) e2[row] = s;
}

// ---------------------------------------------------------------------------
// Kernel 1: init EMA targets (new_avg = avg*DECAY, new_cluster = cs*DECAY)
// and zero the raw-count scratch. 1 thread per (n,k), 1M threads.
// ---------------------------------------------------------------------------
__global__ __launch_bounds__(256) void vq_init_kernel(const float* __restrict__ cs,
                                                      const float* __restrict__ avg,
                                                      float* __restrict__ nc_out,
                                                      float* __restrict__ navg_out,
                                                      float* __restrict__ ntot) {
  int i = blockIdx.x * blockDim.x + threadIdx.x;
  navg_out[i] = avg[i] * DECAY;
  if ((i & (EDIM - 1)) == 0) {
    int n = i >> 8;
    nc_out[n] = cs[n] * DECAY;
    ntot[n]   = 0.f;
  }
}

// ---------------------------------------------------------------------------
// Kernel 2: fused  d = ||e||^2 - 2 z.e  +  argmin over 4096 codes.
// 4 waves / block, 16 tokens / wave, A (16x256 fp32) register-resident.
// B staged via LDS in 128-code chunks using the CDNA5 async-DMA path
// (global_load_async_to_lds_b128, drained with s_wait_asynccnt).
// Two N-tiles per pass -> two independent v_wmma_f32_16x16x4_f32 chains
// sharing each A operand (ILP x2, A-register reuse x2).
// ---------------------------------------------------------------------------
__global__ __launch_bounds__(128) void vq_dist_argmin_kernel(
    const float* __restrict__ z, const float* __restrict__ emb,
    const float* __restrict__ e2, int* __restrict__ idx_out,
    float* __restrict__ idx_f_out) {
  __shared__ float ldsE[CHUNK * LDS_PITCH];

  const int tid  = threadIdx.x;
  const int wave = tid >> 5;
  const int ln   = tid & 31;
  const int m    = ln & 15;     // token row within tile (A) / code col (B,C)
  const int hi   = ln >> 4;     // K-pair selector for 16x16x4 fp32 layout

  const int t0 = blockIdx.x * 64 + wave * 16;   // first token of this wave
  const int g  = t0 + m;                        // global token for A row m
  const int b  = g >> 12;                       // z is [8][256][4096]
  const int tl = g & 4095;
  const float* zbase = z + (size_t)b * 256 * 4096 + tl;

  // LDS byte offset of the staging buffer (flat LDS aperture: offset = addr[31:0])
  const unsigned ldsBase = (unsigned)(uintptr_t)(&ldsE[0]);

  // A operand: lane(m,hi) holds K-pairs {4kk+2hi, 4kk+2hi+1} of token row m
  v2f a[EDIM / 4];
  #pragma unroll
  for (int kk = 0; kk < EDIM / 4; ++kk) {
    int c0 = 4 * kk + 2 * hi;
    a[kk].x = zbase[(size_t)c0 * 4096];
    a[kk].y = zbase[(size_t)(c0 + 1) * 4096];
  }

  float best[8];
  int   bidx[8];
  #pragma unroll
  for (int r = 0; r < 8; ++r) { best[r] = 3.4e38f; bidx[r] = 0; }

  for (int ch = 0; ch < NE / CHUNK; ++ch) {
    __syncthreads();   // LDS safe to overwrite
    // async DMA chunk load: 128 codes x 256 floats, b128 per lane-step,
    // no VGPR data movement, tracked by ASYNCcnt
    const float* src = emb + (size_t)ch * CHUNK * EDIM;
    #pragma unroll 4
    for (int q = tid; q < CHUNK * (EDIM / 4); q += 128) {
      int row = q >> 6, c4 = q & 63;
      const float* gp = src + row * EDIM + c4 * 4;
      unsigned lo = ldsBase + (unsigned)((row * LDS_PITCH + c4 * 4) * 4);
      asm volatile("global_load_async_to_lds_b128 %0, %1, off"
                   :: "v"(lo), "v"(gp) : "memory");
    }
    asm volatile("s_wait_asynccnt 0x0" ::: "memory");
    __syncthreads();   // chunk visible to all waves

    for (int nt = 0; nt < CHUNK / 16; nt += 2) {
      // B operands: lane(m,hi) reads rows K{+2hi,+2hi+1} of codes
      // (nt*16+m) and ((nt+1)*16+m)
      const float* brow0 = &ldsE[(nt * 16 + m) * LDS_PITCH + 2 * hi];
      const float* brow1 = brow0 + 16 * LDS_PITCH;
      v8f acc0, acc1;
      #pragma unroll
      for (int j = 0; j < 8; ++j) { acc0[j] = 0.0f; acc1[j] = 0.0f; }
      #pragma unroll
      for (int kk = 0; kk < EDIM / 4; ++kk) {
        v2f bv0 = *(const v2f*)(brow0 + 4 * kk);
        v2f bv1 = *(const v2f*)(brow1 + 4 * kk);
        acc0 = __builtin_amdgcn_wmma_f32_16x16x4_f32(
            false, a[kk], false, bv0, (short)0, acc0, false, false);
        acc1 = __builtin_amdgcn_wmma_f32_16x16x4_f32(
            false, a[kk], false, bv1, (short)0, acc1, false, false);
      }
      int   n0   = ch * CHUNK + nt * 16 + m;   // code ids at this lane's col
      int   n1   = n0 + 16;
      float e2v0 = e2[n0];
      float e2v1 = e2[n1];
      #pragma unroll
      for (int r = 0; r < 8; ++r) {            // rows r (lo half) / r+8 (hi)
        float d0 = fmaf(-2.0f, acc0[r], e2v0); // ||z||^2 const -> dropped
        float d1 = fmaf(-2.0f, acc1[r], e2v1);
        if (d0 < best[r]) { best[r] = d0; bidx[r] = n0; }
        if (d1 < best[r]) { best[r] = d1; bidx[r] = n1; }
      }
    }
  }

  // cross-column argmin: butterfly inside each 16-lane half (wave32 shfl)
  #pragma unroll
  for (int r = 0; r < 8; ++r) {
    float bv = best[r];
    int   bi = bidx[r];
    #pragma unroll
    for (int s = 1; s < 16; s <<= 1) {
      float ov = __shfl_xor(bv, s, 32);
      int   oi = __shfl_xor(bi, s, 32);
      if (ov < bv || (ov == bv && oi < bi)) { bv = ov; bi = oi; }
    }
    if (m == 0) {
      int t = t0 + r + 8 * hi;     // lane0 -> rows 0..7, lane16 -> rows 8..15
      idx_out[t]   = bi;
      idx_f_out[t] = (float)bi;
    }
  }
}

// ---------------------------------------------------------------------------
// Kernel 3: EMA scatter. One thread per z element (coalesced over z layout),
// float atomics accumulate (1-DECAY)*zf into new_avg, counts into nc / ntot.
// ---------------------------------------------------------------------------
__global__ __launch_bounds__(256) void vq_scatter_kernel(
    const float* __restrict__ z, const int* __restrict__ idx,
    float* __restrict__ navg_out, float* __restrict__ nc_out,
    float* __restrict__ ntot) {
  int i = blockIdx.x * blockDim.x + threadIdx.x;  // linear over [8][256][4096]
  float v = z[i];
  int b  = i >> 20;
  int c  = (i >> 12) & 255;
  int tl = i & 4095;
  int id = idx[b * 4096 + tl];
  atomicAdd(&navg_out[(size_t)id * EDIM + c], (1.0f - DECAY) * v);
  if (c == 0) {
    atomicAdd(&nc_out[id], 1.0f - DECAY);
    atomicAdd(&ntot[id], 1.0f);
  }
}

// ---------------------------------------------------------------------------
// Kernel 4: single-block reduction: n = sum(new_cluster), usage = mean(ntot>0)
// ---------------------------------------------------------------------------
__global__ __launch_bounds__(1024) void vq_reduce_kernel(
    const float* __restrict__ nc, const float* __restrict__ ntot,
    float* __restrict__ nsum_ws, float* __restrict__ usage_out) {
  __shared__ float s1[32], s2[32];
  float a = 0.f, u = 0.f;
  for (int i = threadIdx.x; i < NE; i += 1024) {
    a += nc[i];
    u += (ntot[i] > 0.f) ? 1.f : 0.f;
  }
  #pragma unroll
  for (int s = 16; s >= 1; s >>= 1) { a += __shfl_xor(a, s, 32); u += __shfl_xor(u, s, 32); }
  int w = threadIdx.x >> 5, ln = threadIdx.x & 31;
  if (ln == 0) { s1[w] = a; s2[w] = u; }
  __syncthreads();
  if (w == 0) {
    a = s1[ln]; u = s2[ln];
    #pragma unroll
    for (int s = 16; s >= 1; s >>= 1) { a += __shfl_xor(a, s, 32); u += __shfl_xor(u, s, 32); }
    if (ln == 0) { nsum_ws[0] = a; usage_out[0] = u / (float)NE; }
  }
}

// ---------------------------------------------------------------------------
// Kernel 5: new_embed = new_avg / Laplace-smoothed cluster size
// ---------------------------------------------------------------------------
__global__ __launch_bounds__(256) void vq_embed_kernel(
    const float* __restrict__ navg, const float* __restrict__ nc,
    const float* __restrict__ nsum, float* __restrict__ ne_out) {
  int i = blockIdx.x * blockDim.x + threadIdx.x;  // 1M over [4096][256]
  int n = i >> 8;
  float N  = nsum[0];
  float cs = (nc[n] + EPSV) / (N + (float)NE * EPSV) * N;
  ne_out[i] = navg[i] / cs;
}

// ---------------------------------------------------------------------------
// Kernel 6: zq gather (straight-through => forward value is pure gather),
// written back in b,c,h,w,d layout. Coalesced over output; emb is L2-resident.
// ---------------------------------------------------------------------------
__global__ __launch_bounds__(256) void vq_zq_kernel(
    const float* __restrict__ emb, const int* __restrict__ idx,
    float* __restrict__ zq) {
  int i = blockIdx.x * blockDim.x + threadIdx.x;  // linear over [8][256][4096]
  int b  = i >> 20;
  int c  = (i >> 12) & 255;
  int tl = i & 4095;
  int id = idx[b * 4096 + tl];
  zq[i] = emb[(size_t)id * EDIM + c];
}

// ---------------------------------------------------------------------------
extern "C" void kernel_launch(void* const* d_in, const int* in_sizes, int n_in,
                              void* d_out, int out_size, void* d_ws, size_t ws_size,
                              hipStream_t stream) {
  const float* z   = (const float*)d_in[0];   // [8,256,16,16,16]
  const float* emb = (const float*)d_in[1];   // [4096,256]
  const float* cs  = (const float*)d_in[2];   // [4096]
  const float* avg = (const float*)d_in[3];   // [4096,256]

  // outputs concatenated flat in return order
  float* out    = (float*)d_out;
  float* zq     = out;                         //  8,388,608
  float* idx_f  = zq + 8388608;                //     32,768
  float* nc     = idx_f + 32768;               //      4,096
  float* navg   = nc + 4096;                   //  1,048,576
  float* nemb   = navg + 1048576;              //  1,048,576
  float* usage  = nemb + 1048576;              //          1

  // workspace
  float* e2    = (float*)d_ws;                 // 4096
  int*   idx_i = (int*)(e2 + NE);              // 32768
  float* ntot  = (float*)(idx_i + NTOK);       // 4096
  float* nsum  = ntot + NE;                    // 1

  vq_e2_kernel      <<<NE / 8, 256, 0, stream>>>(emb, e2);
  vq_init_kernel    <<<(NE * EDIM) / 256, 256, 0, stream>>>(cs, avg, nc, navg, ntot);
  vq_dist_argmin_kernel<<<NTOK / 64, 128, 0, stream>>>(z, emb, e2, idx_i, idx_f);
  vq_scatter_kernel <<<(NTOK * EDIM) / 256, 256, 0, stream>>>(z, idx_i, navg, nc, ntot);
  vq_reduce_kernel  <<<1, 1024, 0, stream>>>(nc, ntot, nsum, usage);
  vq_embed_kernel   <<<(NE * EDIM) / 256, 256, 0, stream>>>(navg, nc, nsum, nemb);
  vq_zq_kernel      <<<(NTOK * EDIM) / 256, 256, 0, stream>>>(emb, idx_i, zq);
}